// GNN_81930796138624
// MI455X (gfx1250) — compile-verified
//
#include <hip/hip_runtime.h>

typedef float v2f __attribute__((ext_vector_type(2)));
typedef float v8f __attribute__((ext_vector_type(8)));

// ---------------------------------------------------------------------------
// Kernel 1: zero the workspace accumulators (agg[N,2] ++ pooled[64,2]).
// Must run every launch: graph replays do not re-poison d_ws.
// ---------------------------------------------------------------------------
__global__ void zero_ws_kernel(float* __restrict__ ws, int n) {
    int i = blockIdx.x * blockDim.x + threadIdx.x;
    int stride = gridDim.x * blockDim.x;
    for (; i < n; i += stride) ws[i] = 0.0f;
}

// ---------------------------------------------------------------------------
// Kernel 2: edge phase. Streaming-bound: ~450MB of edge data at 23.3 TB/s.
// Non-temporal loads keep the one-pass edge streams from evicting the
// L2-resident x (2MB) and agg (4MB) working sets.
// ---------------------------------------------------------------------------
__global__ void edge_kernel(const float* __restrict__ x,
                            const float* __restrict__ ea,
                            const int*   __restrict__ rows,
                            const int*   __restrict__ cols,
                            float*       __restrict__ agg,
                            int E) {
    int e = blockIdx.x * blockDim.x + threadIdx.x;
    if (e >= E) return;
    int r  = __builtin_nontemporal_load(rows + e);
    int c  = __builtin_nontemporal_load(cols + e);
    float d  = __builtin_nontemporal_load(ea + 3 * e + 0);
    float b_ = __builtin_nontemporal_load(ea + 3 * e + 2);
    float vi = x[r];
    float vj = x[c];
    float e0 = fabsf((vi / 0.9484139f - (vj - 0.2123214f)) * -1.3248432f)
             + (d - 1.7348461f + b_ + vj) * -0.12084719f;
    float e1 = fabsf((vi - vj * 1.0584362f) * 1.5344211f + 0.45368108f)
             + (vi - vj * 1.0239582f) * 1.931712f + 0.546892f;
    atomicAdd(&agg[2 * c + 0], e0);   // global_atomic_add_f32, L2-resident
    atomicAdd(&agg[2 * c + 1], e1);
}

// ---------------------------------------------------------------------------
// Kernel 3: node phase. pow(const, x) -> exp2f(x * log2(const)).
// batch is sorted, so per-block LDS reduction collapses 1M global atomics on
// 128 addresses down to ~2 per block.
// ---------------------------------------------------------------------------
__global__ void node_kernel(const float* __restrict__ x,
                            const float* __restrict__ agg,
                            const int*   __restrict__ batch,
                            float*       __restrict__ pooled,
                            int N) {
    __shared__ float acc[128];  // 64 batches x 2 channels
    for (int j = threadIdx.x; j < 128; j += blockDim.x) acc[j] = 0.0f;
    __syncthreads();

    int i = blockIdx.x * blockDim.x + threadIdx.x;
    if (i < N) {
        float vn = x[i];
        float s1 = agg[2 * i + 0];
        float s2 = agg[2 * i + 1];
        // log2 constants: log2(0.7660379)=-0.38451248, log2(0.12117091)=-3.0448844
        //                 log2(1.2125463)= 0.27803910, log2(0.1562228) =-2.6782825
        float n1 = (exp2f((s2 / 0.3038425f + s1) * -0.38451248f)
                  + exp2f(s1 * -3.0448844f) / -0.7256157f)
                 * exp2f(vn * 0.2780391f) + 0.12262904f;
        float t  = s2 + (s1 + -3.283101f - vn / 0.79082423f) * 0.31992579f;
        float n1n2 = 0.7872602f - sqrtf(logf(exp2f(t * -2.6782825f) + 1.4462701f));
        int b = batch[i];
        atomicAdd(&acc[2 * b + 0], n1);          // ds_add_f32
        atomicAdd(&acc[2 * b + 1], n1n2 - n1);
    }
    __syncthreads();
    for (int j = threadIdx.x; j < 128; j += blockDim.x) {
        float v = acc[j];
        if (v != 0.0f) atomicAdd(&pooled[j], v); // sorted batch => ~2 hits/block
    }
}

// ---------------------------------------------------------------------------
// Kernel 4: 4-layer 2x2 MLP on pooled[64,2] via chained V_WMMA_F32_16X16X4_F32.
// 4 waves, one 16-row tile each. K padded 2->4, N padded 2->16 with zeros;
// bias preloaded into C. D (row-striped across lanes) is re-laid-out into the
// next A operand (row-per-lane, K in 2 VGPRs) through a per-wave LDS tile.
// ---------------------------------------------------------------------------
__global__ void __launch_bounds__(128) mlp_wmma_kernel(
        const float* __restrict__ pooled,
        const float* __restrict__ W1, const float* __restrict__ b1,
        const float* __restrict__ W2, const float* __restrict__ b2,
        const float* __restrict__ W3, const float* __restrict__ b3,
        const float* __restrict__ W4, const float* __restrict__ b4,
        float* __restrict__ out) {
    __shared__ float stage[4 * 16 * 16];     // one 16x16 f32 tile per wave
    const int lane = threadIdx.x & 31;       // wave32
    const int wave = threadIdx.x >> 5;
    const int n    = lane & 15;              // column for B/C/D layouts
    const int hi   = lane >> 4;              // 0: lanes 0-15, 1: lanes 16-31
    const int row0 = wave * 16;
    float* myStage = &stage[wave * 256];

    // First A operand (16x4, K=2..3 zero-padded): lanes 0-15 carry row M=lane.
    v2f a; a.x = 0.0f; a.y = 0.0f;
    if (hi == 0) {
        a.x = pooled[2 * (row0 + n) + 0];
        a.y = pooled[2 * (row0 + n) + 1];
    }

    const float* Ws[4] = {W1, W2, W3, W4};
    const float* bs[4] = {b1, b2, b3, b4};

#pragma unroll
    for (int l = 0; l < 4; ++l) {
        const float* W  = Ws[l];
        const float* bb = bs[l];

        // B operand (4x16): lanes 0-15 carry column N=n, VGPR v = row K=v.
        // Rows K>=2 (lanes 16-31) and columns N>=2 are zero padding.
        v2f bm; bm.x = 0.0f; bm.y = 0.0f;
        if (hi == 0 && n < 2) { bm.x = W[0 * 2 + n]; bm.y = W[1 * 2 + n]; }

        // C preloaded with bias broadcast down each column.
        v8f c;
        float cinit = (n < 2) ? bb[n] : 0.0f;
#pragma unroll
        for (int r = 0; r < 8; ++r) c[r] = cinit;

        // D = A*B + C  (exact f32 matrix path on CDNA5)
        c = __builtin_amdgcn_wmma_f32_16x16x4_f32(
                false, a, false, bm, (short)0, c, false, false);

        if (l == 3) {
            // Final layer: no relu. o[:,1] lives in column n==1 of D.
            if (n == 1) {
#pragma unroll
                for (int r = 0; r < 8; ++r) {
                    int m = r + hi * 8;
                    out[2 * (row0 + m) + 1] = c[r];
                }
            }
            // o[:,0] is a closed-form function of pooled; column n==0 lanes do it.
            if (n == 0) {
#pragma unroll
                for (int r = 0; r < 8; ++r) {
                    int row = row0 + r + hi * 8;
                    float p1 = pooled[2 * row + 0];
                    float p2 = pooled[2 * row + 1];
                    float o0 = (p2 / -0.18032177f + p1 * 2.2054937f
                              + fabsf(p2 * 0.9565731f + p1 * 0.8225316f))
                               * 0.00046277698f + -0.24634261f;
                    out[2 * row + 0] = o0;
                }
            }
        } else {
            // ReLU(D) -> LDS (row-major 16x16), then reload as next A operand.
#pragma unroll
            for (int r = 0; r < 8; ++r) {
                int m = r + hi * 8;
                myStage[m * 16 + n] = fmaxf(c[r], 0.0f);
            }
            __syncthreads();
            a.x = 0.0f; a.y = 0.0f;
            if (hi == 0) {
                a.x = myStage[n * 16 + 0];
                a.y = myStage[n * 16 + 1];
            }
            __syncthreads();
        }
    }
}

// ---------------------------------------------------------------------------
// Launch: zero -> edges -> nodes -> MLP, all stream-ordered (graph-capturable).
// ---------------------------------------------------------------------------
extern "C" void kernel_launch(void* const* d_in, const int* in_sizes, int n_in,
                              void* d_out, int out_size, void* d_ws, size_t ws_size,
                              hipStream_t stream) {
    const float* x   = (const float*)d_in[0];
    const float* ea  = (const float*)d_in[1];
    const float* W1  = (const float*)d_in[2];
    const float* b1  = (const float*)d_in[3];
    const float* W2  = (const float*)d_in[4];
    const float* b2  = (const float*)d_in[5];
    const float* W3  = (const float*)d_in[6];
    const float* b3  = (const float*)d_in[7];
    const float* W4  = (const float*)d_in[8];
    const float* b4  = (const float*)d_in[9];
    const int* ei    = (const int*)d_in[10];
    const int* batch = (const int*)d_in[11];

    const int N = in_sizes[0];
    const int E = in_sizes[1] / 3;
    const int* rows = ei;
    const int* cols = ei + E;

    float* agg    = (float*)d_ws;            // N x 2 f32
    float* pooled = agg + (size_t)2 * N;     // 64 x 2 f32

    const int zn = 2 * N + 128;
    zero_ws_kernel<<<1024, 256, 0, stream>>>(agg, zn);
    edge_kernel<<<(E + 255) / 256, 256, 0, stream>>>(x, ea, rows, cols, agg, E);
    node_kernel<<<(N + 255) / 256, 256, 0, stream>>>(x, agg, batch, pooled, N);
    mlp_wmma_kernel<<<1, 128, 0, stream>>>(pooled, W1, b1, W2, b2, W3, b3, W4, b4,
                                           (float*)d_out);
}